// Gradient_Net_iqa_21036749816347
// MI455X (gfx1250) — compile-verified
//
#include <hip/hip_runtime.h>

// Sobel gradient magnitude: |conv(x, Kx)| + |conv(x, Ky)| + 1e-5
// Input  : (32, 1, 1024, 1024) f32, SAME zero padding
// Output : (32, 1, 1024, 1024) f32
//
// Memory-bound (AI ~1.6 flop/B). Strategy: CDNA5 async global->LDS DMA for the
// halo tile, LDS-tiled separable stencil, non-temporal 128-bit output stores.

#define IMG_H 1024
#define IMG_W 1024
#define TW 128              // tile width  (outputs)
#define TH 32               // tile height (outputs)
#define LDS_ROWS (TH + 2)   // 34 rows (1 halo row each side)
#define LDS_COLS (TW + 8)   // 136 cols (4 halo cols each side -> 16B aligned chunks)
#define LDS_PITCH_B (LDS_COLS * 4)          // 544 bytes per row
#define CHUNKS_PER_ROW (LDS_COLS / 4)       // 34 x 16B chunks per row
#define NCHUNK (LDS_ROWS * CHUNKS_PER_ROW)  // 1156 chunks per tile
#define NTHREADS 256

typedef float v4f __attribute__((ext_vector_type(4)));

__global__ __launch_bounds__(NTHREADS)
void sobel_mag_kernel(const float* __restrict__ x, float* __restrict__ out) {
    __shared__ alignas(16) float tile[LDS_ROWS][LDS_COLS];

    const int tid = threadIdx.x;
    const int img = blockIdx.z;
    const int tx0 = blockIdx.x * TW;
    const int ty0 = blockIdx.y * TH;

    const float* p_img = x + (size_t)img * (IMG_H * IMG_W);

    // LDS byte offset of the tile: low 32 bits of the generic pointer are the
    // wave-relative LDS address (CDNA5 ISA aperture mapping).
    const unsigned lds_base = (unsigned)(size_t)(void*)&tile[0][0];

    // ---- Stage halo tile into LDS with async DMA (ASYNCcnt path) ----------
    // Each chunk = 16 bytes. Addresses are clamped in-bounds; border tiles
    // zero-patch the out-of-image cells below before the barrier.
    for (int idx = tid; idx < NCHUNK; idx += NTHREADS) {
        const int row = idx / CHUNKS_PER_ROW;
        const int cc  = idx - row * CHUNKS_PER_ROW;
        const int gy  = ty0 - 1 + row;
        const int gx  = tx0 - 4 + cc * 4;
        const int gyc = gy < 0 ? 0 : (gy > IMG_H - 1 ? IMG_H - 1 : gy);
        const int gxc = gx < 0 ? 0 : (gx > IMG_W - 4 ? IMG_W - 4 : gx);
        const unsigned voff   = (unsigned)((gyc * IMG_W + gxc) * 4);
        const unsigned ldsoff = lds_base + (unsigned)(row * LDS_PITCH_B + cc * 16);
        asm volatile("global_load_async_to_lds_b128 %0, %1, %2"
                     :: "v"(ldsoff), "v"(voff), "s"(p_img)
                     : "memory");
    }
    asm volatile("s_wait_asynccnt 0" ::: "memory");

    // ---- Zero-patch out-of-image cells (border tiles only) ----------------
    const bool border = (tx0 == 0) | (ty0 == 0) |
                        (tx0 + TW == IMG_W) | (ty0 + TH == IMG_H);
    if (border) {
        for (int idx = tid; idx < NCHUNK; idx += NTHREADS) {
            const int row = idx / CHUNKS_PER_ROW;
            const int cc  = idx - row * CHUNKS_PER_ROW;
            const int gy  = ty0 - 1 + row;
            const int gx0 = tx0 - 4 + cc * 4;
            const bool rowOOB = (gy < 0) | (gy >= IMG_H);
            #pragma unroll
            for (int i = 0; i < 4; ++i) {
                const int gx = gx0 + i;
                if (rowOOB | (gx < 0) | (gx >= IMG_W))
                    tile[row][cc * 4 + i] = 0.0f;
            }
        }
    }
    __syncthreads();

    // ---- Compute: each thread -> 4x4 output patch from a 6x6 LDS window ---
    const int cx = tid & 31;   // column group: 32 groups of 4 cols (coalesced)
    const int ry = tid >> 5;   // row group:     8 groups of 4 rows
    const int c0 = cx * 4 + 3; // left edge of 6-wide window in LDS
    const int r0 = ry * 4;     // top edge of 6-tall window in LDS

    float w[6][6];
    #pragma unroll
    for (int j = 0; j < 6; ++j)
        #pragma unroll
        for (int i = 0; i < 6; ++i)
            w[j][i] = tile[r0 + j][c0 + i];

    const size_t obase = ((size_t)img * IMG_H + (size_t)(ty0 + ry * 4)) * IMG_W
                       + (size_t)(tx0 + cx * 4);

    #pragma unroll
    for (int j = 0; j < 4; ++j) {
        // Separable: vertical temps over the 6-wide window.
        // v1 = [1,2,1]  vertical smooth (for grad_x)
        // v2 = in[y+1]-in[y-1]          (for grad_y)
        float v1[6], v2[6];
        #pragma unroll
        for (int i = 0; i < 6; ++i) {
            v1[i] = w[j][i] + 2.0f * w[j + 1][i] + w[j + 2][i];
            v2[i] = w[j + 2][i] - w[j][i];
        }
        float r[4];
        #pragma unroll
        for (int i = 0; i < 4; ++i) {
            const float gx = v1[i + 2] - v1[i];                        // [-1,0,1] horiz
            const float gy = v2[i] + 2.0f * v2[i + 1] + v2[i + 2];     // [1,2,1]  horiz
            r[i] = fabsf(gx) + fabsf(gy) + 1e-5f;
        }
        v4f o;
        o.x = r[0]; o.y = r[1]; o.z = r[2]; o.w = r[3];
        // Streaming output: non-temporal so the 128MB input stays L2-resident.
        __builtin_nontemporal_store(o, (v4f*)(out + obase + (size_t)j * IMG_W));
    }
}

extern "C" void kernel_launch(void* const* d_in, const int* in_sizes, int n_in,
                              void* d_out, int out_size, void* d_ws, size_t ws_size,
                              hipStream_t stream) {
    const float* x = (const float*)d_in[0];
    float* out = (float*)d_out;
    const int nimg = in_sizes[0] / (IMG_H * IMG_W);   // 32 images of 1024x1024
    dim3 grid(IMG_W / TW, IMG_H / TH, nimg);          // (8, 32, 32)
    sobel_mag_kernel<<<grid, NTHREADS, 0, stream>>>(x, out);
}